// NONLocalBlock1D_mutual_43851616092414
// MI455X (gfx1250) — compile-verified
//
#include <hip/hip_runtime.h>

// Problem constants (match reference)
constexpr int B  = 8;
constexpr int C  = 512;
constexpr int P  = 4096;
constexpr int IC = 256;      // C/2
constexpr float EPS = 1e-5f;

typedef __attribute__((ext_vector_type(2))) float v2f;
typedef __attribute__((ext_vector_type(8))) float v8f;

// D = A(16x4) * B(4x16) + C, fp32, wave32.
// A layout: lanes 0-15 hold M=lane, VGPR0=K0,VGPR1=K1 ; lanes 16-31 M=lane-16, K2,K3
// B layout: lanes 0-15 hold N=lane, VGPR0=K0,VGPR1=K1 ; lanes 16-31 N=lane-16, K2,K3
// C/D layout: lanes 0-15: comp r = (M=r, N=lane); lanes 16-31: comp r = (M=8+r, N=lane-16)
__device__ __forceinline__ v8f wmma4(v2f a, v2f b, v8f c) {
  return __builtin_amdgcn_wmma_f32_16x16x4_f32(false, a, false, b, (short)0, c, false, false);
}

// q/k intermediate layout: pair-interleaved (B, IC/2, P, 2):
//   element (b, o, p) lives at  ((b*(IC/2) + o/2)*P + p)*2 + (o&1)
// so a WMMA K-pair fragment (K=o, K=o+1) for a lane is ONE aligned 8-byte load,
// and lanes 0..15 (consecutive p) cover 128 contiguous bytes.

// ---------------------------------------------------------------------------
// k0: xsum[b,c] = sum_p x[b,c,p]
__global__ __launch_bounds__(256) void xsum_kernel(const float* __restrict__ x,
                                                   float* __restrict__ xsum) {
  const int row = blockIdx.x;                      // b*C + c
  const float4* x4 = (const float4*)(x + (size_t)row * P);
  float s = 0.f;
  for (int k = threadIdx.x; k < P / 4; k += 256) {
    float4 v = x4[k];
    s += v.x + v.y + v.z + v.w;
  }
  __shared__ float red[256];
  red[threadIdx.x] = s;
  __syncthreads();
  for (int off = 128; off > 0; off >>= 1) {
    if ((int)threadIdx.x < off) red[threadIdx.x] += red[threadIdx.x + off];
    __syncthreads();
  }
  if (threadIdx.x == 0) xsum[row] = red[0];
}

// ---------------------------------------------------------------------------
// k1: vsum (in LDS) -> dcoef[b,c] = u1-u2, u2p[b,c] = u2 + b_W[c]
__global__ __launch_bounds__(256) void small_coefs(const float* __restrict__ xsum,
                                                   const float* __restrict__ w_g,
                                                   const float* __restrict__ b_g,
                                                   const float* __restrict__ w_W,
                                                   const float* __restrict__ b_W,
                                                   float* __restrict__ dcoef,
                                                   float* __restrict__ u2p) {
  __shared__ float svs[B * IC];
  const int tid = threadIdx.x;
  for (int idx = tid; idx < B * IC; idx += 256) {
    const int b = idx / IC, o = idx % IC;
    const int bs = (b + B - 1) & (B - 1);          // rolled batch
    const float* xs = xsum + bs * C;
    const float* wr = w_g + (size_t)o * C;
    float acc = 0.f;
    for (int c = 0; c < C; ++c) acc += wr[c] * xs[c];
    svs[idx] = acc + (float)P * b_g[o];
  }
  __syncthreads();
  for (int idx = tid; idx < B * C; idx += 256) {
    const int b = idx / C, c = idx % C;
    const float* wr = w_W + (size_t)c * (2 * IC);
    const float* vs = svs + b * IC;
    float s1 = 0.f, s2 = 0.f;
    for (int o = 0; o < IC; ++o) {
      s1 += wr[o] * vs[o];
      s2 += wr[IC + o] * vs[o];
    }
    dcoef[idx] = s1 - s2;
    u2p[idx]   = s2 + b_W[c];
  }
}

// ---------------------------------------------------------------------------
// k2: projections into pair-interleaved layout.
//   qt(b,o,p) = sum_c w_theta[o,c]*x[b,c,p] + b_theta[o]
//   kn(b,o,p) = sum_c w_phi[o,c]*xr[b,c,p]  + b_phi[o]
// Tile: M = out-channel (16), N = point (16), K = c. float2 stores, coalesced.
__global__ __launch_bounds__(256) void proj_qk(const float* __restrict__ x,
                                               const float* __restrict__ w_theta,
                                               const float* __restrict__ b_theta,
                                               const float* __restrict__ w_phi,
                                               const float* __restrict__ b_phi,
                                               float* __restrict__ qt,
                                               float* __restrict__ kn) {
  const int t    = blockIdx.x * 8 + (threadIdx.x >> 5);   // global tile id
  const int mat  = t >> 15;                               // 0 = q, 1 = k
  const int rem  = t & 32767;
  const int b    = rem >> 12;                             // 4096 tiles per (mat,b)
  const int rem2 = rem & 4095;
  const int o0   = (rem2 >> 8) << 4;                      // 16 o-tiles
  const int p0   = (rem2 & 255) << 4;                     // 256 p-tiles
  const int lane  = threadIdx.x & 31;
  const int mlane = lane & 15;
  const int kk    = (lane < 16) ? 0 : 2;

  const int bs = mat ? ((b + B - 1) & (B - 1)) : b;       // k/v use rolled batch
  const float* w    = mat ? w_phi : w_theta;
  const float* bias = mat ? b_phi : b_theta;
  const float* xb   = x + (size_t)bs * C * P;
  float* outp = (mat ? kn : qt) + (size_t)b * IC * P;

  const float* pa = w  + (size_t)(o0 + mlane) * C + kk;   // A(M=o,K=c): weights (b64)
  const float* pb = xb + (size_t)kk * P + p0 + mlane;     // B(K=c,N=p): x, coalesced

  v8f acc0 = {}, acc1 = {};
  for (int c = 0; c < C; c += 8) {
    v2f a0 = *(const v2f*)(pa);
    v2f b0; b0.x = pb[0]; b0.y = pb[P];
    acc0 = wmma4(a0, b0, acc0);
    v2f a1 = *(const v2f*)(pa + 4);
    v2f b1; b1.x = pb[4 * (size_t)P]; b1.y = pb[5 * (size_t)P];
    acc1 = wmma4(a1, b1, acc1);
    pa += 8;
    pb += 8 * (size_t)P;
  }
  const int half = (lane < 16) ? 0 : 8;
  const int pairbase = (o0 + half) >> 1;                  // pair-row index
#pragma unroll
  for (int r = 0; r < 4; ++r) {
    v2f vv;
    vv.x = acc0[2 * r]     + acc1[2 * r]     + bias[o0 + half + 2 * r];
    vv.y = acc0[2 * r + 1] + acc1[2 * r + 1] + bias[o0 + half + 2 * r + 1];
    *(v2f*)(outp + ((size_t)(pairbase + r) * P + p0 + mlane) * 2) = vv;  // b64, coalesced
  }
}

// ---------------------------------------------------------------------------
// k3: per-row logsumexp of f[b,i,:], f[i,j] = sum_o q(b,o,i)*k(b,o,j).
// One block (8 waves) per (b, i-tile); waves split the 256 j-tiles.
// Pair-interleaved q/k: every fragment is one aligned b64 load.
__global__ __launch_bounds__(256) void attn_rowstats(const float* __restrict__ qt,
                                                     const float* __restrict__ kn,
                                                     float* __restrict__ Lrow) {
  const int blk = blockIdx.x;                 // b*(P/16) + it
  const int b   = blk >> 8;
  const int i0  = (blk & 255) << 4;
  const int tid = threadIdx.x;
  const int lane = tid & 31, wave = tid >> 5;
  const int mlane = lane & 15;
  const int kh    = (lane < 16) ? 0 : 1;      // K offset in pairs (K+2,K+3 half)

  const float* qb = qt + (size_t)b * IC * P;
  const float* kb = kn + (size_t)b * IC * P;
  const float* pa_base = qb + ((size_t)kh * P + i0 + mlane) * 2;  // A(M=i,K=o)

  float lmax[8], lsum[8];
#pragma unroll
  for (int r = 0; r < 8; ++r) { lmax[r] = -1e30f; lsum[r] = 0.f; }

  for (int jt = wave; jt < P / 16; jt += 8) {
    const int j0 = jt << 4;
    const float* pa = pa_base;
    const float* pb = kb + ((size_t)kh * P + j0 + mlane) * 2;     // B(K=o,N=j)
    v8f acc0 = {}, acc1 = {};
    for (int o = 0; o < IC; o += 8) {
      v2f a0 = *(const v2f*)(pa);
      v2f b0 = *(const v2f*)(pb);
      acc0 = wmma4(a0, b0, acc0);
      v2f a1 = *(const v2f*)(pa + 4 * (size_t)P);
      v2f b1 = *(const v2f*)(pb + 4 * (size_t)P);
      acc1 = wmma4(a1, b1, acc1);
      pa += 8 * (size_t)P;                     // 4 pair-rows forward
      pb += 8 * (size_t)P;
    }
#pragma unroll
    for (int r = 0; r < 8; ++r) {              // online max/sum
      const float v  = acc0[r] + acc1[r];
      const float nm = fmaxf(lmax[r], v);
      lsum[r] = lsum[r] * __expf(lmax[r] - nm) + __expf(v - nm);
      lmax[r] = nm;
    }
  }
  // combine across the 16 lanes of each half (xor masks stay within halves)
#pragma unroll
  for (int off = 1; off < 16; off <<= 1) {
#pragma unroll
    for (int r = 0; r < 8; ++r) {
      const float om = __shfl_xor(lmax[r], off, 32);
      const float os = __shfl_xor(lsum[r], off, 32);
      const float nm = fmaxf(lmax[r], om);
      lsum[r] = lsum[r] * __expf(lmax[r] - nm) + os * __expf(om - nm);
      lmax[r] = nm;
    }
  }
  __shared__ float sM[8][16], sS[8][16];
  if (lane == 0 || lane == 16) {
    const int rb = (lane == 0) ? 0 : 8;
#pragma unroll
    for (int r = 0; r < 8; ++r) { sM[wave][rb + r] = lmax[r]; sS[wave][rb + r] = lsum[r]; }
  }
  __syncthreads();
  if (tid < 16) {
    float M = sM[0][tid], S = sS[0][tid];
#pragma unroll
    for (int w = 1; w < 8; ++w) {
      const float om = sM[w][tid], os = sS[w][tid];
      const float nm = fmaxf(M, om);
      S = S * __expf(M - nm) + os * __expf(om - nm);
      M = nm;
    }
    Lrow[b * P + i0 + tid] = M + __logf(S);
  }
}

// ---------------------------------------------------------------------------
// k4: m[b,j] = (1/P) * sum_i exp(f[i,j] - L[b,i]).
// Transposed tiles: M=j (k rows), N=i (q cols). One wave owns one j-tile -> deterministic.
__global__ __launch_bounds__(256) void attn_colsum(const float* __restrict__ qt,
                                                   const float* __restrict__ kn,
                                                   const float* __restrict__ Lrow,
                                                   float* __restrict__ mbuf) {
  const int wgid = blockIdx.x * 8 + (threadIdx.x >> 5);  // 0..B*(P/16)-1
  const int b  = wgid >> 8;
  const int j0 = (wgid & 255) << 4;
  const int lane  = threadIdx.x & 31;
  const int mlane = lane & 15;
  const int kh    = (lane < 16) ? 0 : 1;

  const float* qb = qt + (size_t)b * IC * P;
  const float* kb = kn + (size_t)b * IC * P;
  const float* pa_base = kb + ((size_t)kh * P + j0 + mlane) * 2;  // A(M=j,K=o)
  const float* Lb = Lrow + b * P;

  float macc[8];
#pragma unroll
  for (int r = 0; r < 8; ++r) macc[r] = 0.f;

  for (int it = 0; it < P / 16; ++it) {
    const int i0 = it << 4;
    const float Lv = Lb[i0 + mlane];                        // per column i
    const float* pa = pa_base;
    const float* pb = qb + ((size_t)kh * P + i0 + mlane) * 2;  // B(K=o,N=i)
    v8f acc0 = {}, acc1 = {};
    for (int o = 0; o < IC; o += 8) {
      v2f a0 = *(const v2f*)(pa);
      v2f b0 = *(const v2f*)(pb);
      acc0 = wmma4(a0, b0, acc0);
      v2f a1 = *(const v2f*)(pa + 4 * (size_t)P);
      v2f b1 = *(const v2f*)(pb + 4 * (size_t)P);
      acc1 = wmma4(a1, b1, acc1);
      pa += 8 * (size_t)P;
      pb += 8 * (size_t)P;
    }
#pragma unroll
    for (int r = 0; r < 8; ++r) macc[r] += __expf((acc0[r] + acc1[r]) - Lv);
  }
  // sum over the 16 columns (lanes) of each half
#pragma unroll
  for (int off = 1; off < 16; off <<= 1) {
#pragma unroll
    for (int r = 0; r < 8; ++r) macc[r] += __shfl_xor(macc[r], off, 32);
  }
  const float invP = 1.0f / (float)P;
  if (lane == 0) {
#pragma unroll
    for (int r = 0; r < 8; ++r) mbuf[b * P + j0 + r] = macc[r] * invP;
  } else if (lane == 16) {
#pragma unroll
    for (int r = 0; r < 8; ++r) mbuf[b * P + j0 + 8 + r] = macc[r] * invP;
  }
}

// ---------------------------------------------------------------------------
// k5: per-batch mean(m) and mean(m^2)
__global__ __launch_bounds__(256) void mstats(const float* __restrict__ mbuf,
                                              float* __restrict__ mbar,
                                              float* __restrict__ m2bar) {
  const int b = blockIdx.x;
  const float* mr = mbuf + (size_t)b * P;
  float s1 = 0.f, s2 = 0.f;
  for (int j = threadIdx.x; j < P; j += 256) {
    const float v = mr[j];
    s1 += v;
    s2 += v * v;
  }
  __shared__ float r1[256], r2[256];
  r1[threadIdx.x] = s1;
  r2[threadIdx.x] = s2;
  __syncthreads();
  for (int off = 128; off > 0; off >>= 1) {
    if ((int)threadIdx.x < off) {
      r1[threadIdx.x] += r1[threadIdx.x + off];
      r2[threadIdx.x] += r2[threadIdx.x + off];
    }
    __syncthreads();
  }
  if (threadIdx.x == 0) {
    mbar[b]  = r1[0] / (float)P;
    m2bar[b] = r2[0] / (float)P;
  }
}

// ---------------------------------------------------------------------------
// k6: BN stats over (b,p) per channel -> affine coefs A0/A1 per (b,c)
__global__ __launch_bounds__(256) void bn_coefs(const float* __restrict__ dcoef,
                                                const float* __restrict__ u2p,
                                                const float* __restrict__ mbar,
                                                const float* __restrict__ m2bar,
                                                const float* __restrict__ gamma,
                                                const float* __restrict__ beta,
                                                float* __restrict__ A0,
                                                float* __restrict__ A1) {
  const int c = blockIdx.x * 256 + threadIdx.x;
  if (c >= C) return;
  float mu = 0.f, e2 = 0.f;
  for (int b = 0; b < B; ++b) {
    const float u = u2p[b * C + c], dd = dcoef[b * C + c];
    const float mb = mbar[b], m2 = m2bar[b];
    mu += u + mb * dd;
    e2 += u * u + 2.f * u * dd * mb + dd * dd * m2;
  }
  mu /= (float)B;
  e2 /= (float)B;
  const float var = e2 - mu * mu;
  const float rs  = rsqrtf(var + EPS);
  const float g = gamma[c], be = beta[c];
  for (int b = 0; b < B; ++b) {
    const float u = u2p[b * C + c], dd = dcoef[b * C + c];
    A1[b * C + c] = g * rs * dd;
    A0[b * C + c] = g * rs * (u - mu) + be;
  }
}

// ---------------------------------------------------------------------------
// k7: out[b,c,p] = A0[b,c] + A1[b,c]*m[b,p] + x[b,c,p]
__global__ __launch_bounds__(256) void final_kernel(const float* __restrict__ x,
                                                    const float* __restrict__ mbuf,
                                                    const float* __restrict__ A0,
                                                    const float* __restrict__ A1,
                                                    float* __restrict__ out) {
  const int row = blockIdx.x;                // b*C + c
  const int b = row / C;
  const float a0 = A0[row], a1 = A1[row];
  const float4* x4 = (const float4*)(x + (size_t)row * P);
  const float4* m4 = (const float4*)(mbuf + (size_t)b * P);
  float4* o4 = (float4*)(out + (size_t)row * P);
  for (int k = threadIdx.x; k < P / 4; k += 256) {
    const float4 xv = x4[k];
    const float4 mv = m4[k];
    float4 ov;
    ov.x = a0 + a1 * mv.x + xv.x;
    ov.y = a0 + a1 * mv.y + xv.y;
    ov.z = a0 + a1 * mv.z + xv.z;
    ov.w = a0 + a1 * mv.w + xv.w;
    o4[k] = ov;
  }
}

// ---------------------------------------------------------------------------
extern "C" void kernel_launch(void* const* d_in, const int* in_sizes, int n_in,
                              void* d_out, int out_size, void* d_ws, size_t ws_size,
                              hipStream_t stream) {
  (void)in_sizes; (void)n_in; (void)out_size; (void)ws_size;
  const float* x       = (const float*)d_in[0];
  const float* w_theta = (const float*)d_in[1];
  const float* b_theta = (const float*)d_in[2];
  const float* w_phi   = (const float*)d_in[3];
  const float* b_phi   = (const float*)d_in[4];
  const float* w_g     = (const float*)d_in[5];
  const float* b_g     = (const float*)d_in[6];
  const float* w_W     = (const float*)d_in[7];
  const float* b_W     = (const float*)d_in[8];
  const float* gamma   = (const float*)d_in[9];
  const float* beta    = (const float*)d_in[10];
  float* out = (float*)d_out;

  // q/k tiles live in d_out (overwritten by final_kernel at the very end):
  float* qt = out;                               // B*IC*P floats (pair-interleaved)
  float* kn = out + (size_t)B * IC * P;          // B*IC*P floats (pair-interleaved)
  // small scratch in d_ws:
  float* ws    = (float*)d_ws;
  float* xsum  = ws; ws += B * C;
  float* dcoef = ws; ws += B * C;
  float* u2p   = ws; ws += B * C;
  float* Lrow  = ws; ws += B * P;
  float* mbuf  = ws; ws += B * P;
  float* mbar  = ws; ws += B;
  float* m2bar = ws; ws += B;
  float* A0    = ws; ws += B * C;
  float* A1    = ws; ws += B * C;

  xsum_kernel<<<B * C, 256, 0, stream>>>(x, xsum);
  small_coefs<<<1, 256, 0, stream>>>(xsum, w_g, b_g, w_W, b_W, dcoef, u2p);
  proj_qk<<<(2 * B * (IC / 16) * (P / 16)) / 8, 256, 0, stream>>>(
      x, w_theta, b_theta, w_phi, b_phi, qt, kn);
  attn_rowstats<<<B * (P / 16), 256, 0, stream>>>(qt, kn, Lrow);
  attn_colsum<<<B * (P / 16) / 8, 256, 0, stream>>>(qt, kn, Lrow, mbuf);
  mstats<<<B, 256, 0, stream>>>(mbuf, mbar, m2bar);
  bn_coefs<<<(C + 255) / 256, 256, 0, stream>>>(dcoef, u2p, mbar, m2bar, gamma, beta, A0, A1);
  final_kernel<<<B * C, 256, 0, stream>>>(x, mbuf, A0, A1, out);
}